// DistributedInverseRealSHT_63848983822762
// MI455X (gfx1250) — compile-verified
//
#include <hip/hip_runtime.h>
#include <hip/hip_bf16.h>

// ---------------------------------------------------------------------------
// Distributed inverse real SHT for MI455X (gfx1250), fp32 via V_WMMA_F32_16X16X4_F32.
//
//   k_pack_x   : x(c,l,m,r) -> Xp[m][l][rho]  (LDS-tiled transpose)
//   k_prep     : zero Xp's 3 pad rows per m and Y's 2 pad columns per row
//   k_build_w  : pair-interleaved irfft twiddles W2p[t/2][j][2]
//   k_legendre : per-m GEMM (256 x 364pad) x (364pad x 361) -> Y  (WMMA f32, branchless loop)
//   k_ifft     : GEMM (46208 x 724pad) x (724pad x 720) -> out    (WMMA f32, b64 loads)
//
// Workspace: Xp 134.6MB + Y 133.8MB + W2p 2.1MB  (~271MB in d_ws).
// ---------------------------------------------------------------------------

#define NLAT 361
#define NLON 720
#define LMAX 361
#define LPAD 364              // stage-1 contraction dim padded to mult of 4 (zero rows)
#define MMAX 361
#define NC   128
#define NR   256              // rows of stage-1 GEMM: (c, re/im) = 128*2
#define KPAD 724              // stage-2 contraction dim 2*MMAX=722 padded to mult of 4
#define JPAD 736              // W2 column padding (720 -> 23*32)
#define ROWS2 (NC * NLAT)     // 46208 rows of stage-2 GEMM

typedef __attribute__((ext_vector_type(2))) float v2f;
typedef __attribute__((ext_vector_type(8))) float v8f;

__device__ __forceinline__ v8f wmma_f32(v2f a, v2f b, v8f c) {
  // 8 args: (neg_a, A, neg_b, B, c_mod, C, reuse_a, reuse_b)
  return __builtin_amdgcn_wmma_f32_16x16x4_f32(false, a, false, b, (short)0, c,
                                               false, false);
}

// ---------------------------------------------------------------------------
// Pack x(B=1, C, L, M, 2) -> Xp[m][l][rho], rho = 2*c + r.  One l per block,
// 32 m per block. LDS tile keeps both global sides coalesced.
// ---------------------------------------------------------------------------
__global__ void __launch_bounds__(256)
k_pack_x(const float* __restrict__ x, float* __restrict__ Xp) {
  __shared__ float tile[32][258];
  const int l   = blockIdx.x;
  const int m0  = blockIdx.y * 32;
  const int tid = threadIdx.x;

  for (int t = 0; t < 32; ++t) {
    int flat = t * 256 + tid;
    int c    = flat >> 6;        // 0..127
    int q    = flat & 63;
    int mloc = q >> 1;
    int r    = q & 1;
    int m    = m0 + mloc;
    if (m < MMAX)
      tile[mloc][2 * c + r] = x[((size_t)(c * LMAX + l) * MMAX + m) * 2 + r];
  }
  __syncthreads();
  for (int t = 0; t < 32; ++t) {
    int m = m0 + t;
    if (m < MMAX)
      Xp[((size_t)m * LPAD + l) * NR + tid] = tile[t][tid];
  }
}

// ---------------------------------------------------------------------------
// Zero Xp pad rows (l = 361..363 for each m) and Y pad columns (t = 722,723).
// ---------------------------------------------------------------------------
#define XPAD_N (MMAX * (LPAD - LMAX) * NR)   // 361*3*256 = 277248
#define YPAD_N (ROWS2 * 2)                   // 92416
__global__ void k_prep(float* __restrict__ Xp, float* __restrict__ Y) {
  int idx = blockIdx.x * blockDim.x + threadIdx.x;
  if (idx < XPAD_N) {
    int m   = idx / ((LPAD - LMAX) * NR);
    int rem = idx - m * ((LPAD - LMAX) * NR);
    Xp[((size_t)m * LPAD + LMAX) * NR + rem] = 0.f;
  } else if (idx < XPAD_N + YPAD_N) {
    int k   = idx - XPAD_N;
    Y[(size_t)(k >> 1) * KPAD + 722 + (k & 1)] = 0.f;
  }
}

// ---------------------------------------------------------------------------
// Pair-interleaved twiddles: W2p[tp][j][2] with tp = t/2 = m:
//   [0]:  alpha_m * cos(2*pi*m*j/720)
//   [1]: -alpha_m * sin(2*pi*m*j/720)   (0 for m=0, m=360)
//   alpha_m = 1 for m in {0,360}, else 2.  Zero outside (m<361, j<720).
// ---------------------------------------------------------------------------
__global__ void k_build_w(float* __restrict__ W2p) {
  int idx = blockIdx.x * blockDim.x + threadIdx.x;
  if (idx >= (KPAD / 2) * JPAD) return;
  int tp = idx / JPAD;          // = m when tp < 361
  int j  = idx - tp * JPAD;
  float v0 = 0.f, v1 = 0.f;
  if (tp < MMAX && j < NLON) {
    const double PI = 3.14159265358979323846;
    double th = (2.0 * PI * (double)((tp * j) % NLON)) / (double)NLON;
    double s, c;
    sincos(th, &s, &c);
    bool edge = (tp == 0) || (tp == NLON / 2);
    double alpha = edge ? 1.0 : 2.0;
    v0 = (float)(alpha * c);
    v1 = edge ? 0.f : (float)(-alpha * s);
  }
  W2p[(size_t)idx * 2 + 0] = v0;
  W2p[(size_t)idx * 2 + 1] = v1;
}

// ---------------------------------------------------------------------------
// Stage 1: for each m, Y[(c,k)][2m+r] = sum_l Xp[m][l][rho] * pct[m][l][k].
// 8 waves/block, each wave owns a 32x32 tile (2x2 WMMA accumulators).
// Branchless inner loop: A rows are zero-padded to LPAD; B row/col indices
// are clamped (A==0 on pad rows kills any clamped-B garbage; out-of-range
// columns are discarded by the guarded stores).
// ---------------------------------------------------------------------------
__global__ void __launch_bounds__(256)
k_legendre(const float* __restrict__ Xp, const float* __restrict__ pct,
           float* __restrict__ Y) {
  const int tid  = threadIdx.x;
  const int lane = tid & 31;
  const int wv   = tid >> 5;
  const int half = lane >> 4;
  const int lr   = lane & 15;
  const int m    = blockIdx.x;
  const int k0   = blockIdx.y * 32;
  const int r0   = wv * 32;

  const float* __restrict__ Xm = Xp  + (size_t)m * LPAD * NR;
  const float* __restrict__ Pm = pct + (size_t)m * LMAX * NLAT;

  v8f c00 = {}, c01 = {}, c10 = {}, c11 = {};

  const int ra0  = r0 + lr, ra1 = r0 + 16 + lr;
  const int cb0  = k0 + lr, cb1 = k0 + 16 + lr;
  const bool okb0 = cb0 < NLAT, okb1 = cb1 < NLAT;
  const int cb0c = okb0 ? cb0 : (NLAT - 1);   // clamped load columns
  const int cb1c = okb1 ? cb1 : (NLAT - 1);

#pragma unroll 2
  for (int l0 = 0; l0 < LPAD; l0 += 4) {
    int la  = l0 + 2 * half;
    int lb0 = la < (LMAX - 1) ? la : (LMAX - 1);       // clamp for B only
    int lb1 = (la + 1) < (LMAX - 1) ? (la + 1) : (LMAX - 1);
    v2f a0, a1, b0, b1;
    a0.x = Xm[(size_t)la * NR + ra0];
    a0.y = Xm[(size_t)(la + 1) * NR + ra0];
    a1.x = Xm[(size_t)la * NR + ra1];
    a1.y = Xm[(size_t)(la + 1) * NR + ra1];
    b0.x = Pm[(size_t)lb0 * NLAT + cb0c];
    b0.y = Pm[(size_t)lb1 * NLAT + cb0c];
    b1.x = Pm[(size_t)lb0 * NLAT + cb1c];
    b1.y = Pm[(size_t)lb1 * NLAT + cb1c];
    c00 = wmma_f32(a0, b0, c00);
    c01 = wmma_f32(a0, b1, c01);
    c10 = wmma_f32(a1, b0, c10);
    c11 = wmma_f32(a1, b1, c11);
  }

  // D layout: VGPR v -> M = v + 8*half, N = lr.
#pragma unroll
  for (int v = 0; v < 8; ++v) {
    int Mv   = v + 8 * half;
    int rho0 = r0 + Mv;
    int rho1 = r0 + 16 + Mv;
    if (okb0) {
      Y[((size_t)(rho0 >> 1) * NLAT + cb0) * KPAD + 2 * m + (rho0 & 1)] = c00[v];
      Y[((size_t)(rho1 >> 1) * NLAT + cb0) * KPAD + 2 * m + (rho1 & 1)] = c10[v];
    }
    if (okb1) {
      Y[((size_t)(rho0 >> 1) * NLAT + cb1) * KPAD + 2 * m + (rho0 & 1)] = c01[v];
      Y[((size_t)(rho1 >> 1) * NLAT + cb1) * KPAD + 2 * m + (rho1 & 1)] = c11[v];
    }
  }
}

// ---------------------------------------------------------------------------
// Stage 2: out[(c,k)][j] = sum_t Y[(c,k)][t] * W2[t][j].  46208 x 722 x 720.
// All inner-loop loads are unconditional aligned b64; stores coalesced.
// ---------------------------------------------------------------------------
__global__ void __launch_bounds__(256)
k_ifft(const float* __restrict__ Y, const float* __restrict__ W2p,
       float* __restrict__ out) {
  const int tid  = threadIdx.x;
  const int lane = tid & 31;
  const int wv   = tid >> 5;
  const int half = lane >> 4;
  const int lr   = lane & 15;

  const int jt      = blockIdx.x;
  const int rowTile = blockIdx.y * 8 + wv;   // uniform per wave
  if (rowTile >= ROWS2 / 32) return;         // whole wave exits: EXEC stays full

  const int j0  = jt * 32;
  const int cb0 = j0 + lr, cb1 = j0 + 16 + lr;

  const float* __restrict__ Ya = Y + (size_t)(rowTile * 32 + lr) * KPAD;
  const float* __restrict__ Yb = Y + (size_t)(rowTile * 32 + 16 + lr) * KPAD;

  v8f c00 = {}, c01 = {}, c10 = {}, c11 = {};

#pragma unroll 2
  for (int t0 = 0; t0 < KPAD; t0 += 4) {
    int ta = t0 + 2 * half;                  // even -> 8B aligned
    int tp = ta >> 1;
    v2f a0 = *(const v2f*)(Ya + ta);
    v2f a1 = *(const v2f*)(Yb + ta);
    v2f b0 = *(const v2f*)(W2p + ((size_t)tp * JPAD + cb0) * 2);
    v2f b1 = *(const v2f*)(W2p + ((size_t)tp * JPAD + cb1) * 2);
    c00 = wmma_f32(a0, b0, c00);
    c01 = wmma_f32(a0, b1, c01);
    c10 = wmma_f32(a1, b0, c10);
    c11 = wmma_f32(a1, b1, c11);
  }

  const bool ok0 = cb0 < NLON, ok1 = cb1 < NLON;
#pragma unroll
  for (int v = 0; v < 8; ++v) {
    int Mv = v + 8 * half;
    int ra = rowTile * 32 + Mv;
    int rb = rowTile * 32 + 16 + Mv;
    if (ok0) {
      out[(size_t)ra * NLON + cb0] = c00[v];
      out[(size_t)rb * NLON + cb0] = c10[v];
    }
    if (ok1) {
      out[(size_t)ra * NLON + cb1] = c01[v];
      out[(size_t)rb * NLON + cb1] = c11[v];
    }
  }
}

// ---------------------------------------------------------------------------
extern "C" void kernel_launch(void* const* d_in, const int* in_sizes, int n_in,
                              void* d_out, int out_size, void* d_ws, size_t ws_size,
                              hipStream_t stream) {
  const float* x   = (const float*)d_in[0];   // (1,128,361,361,2)
  const float* pct = (const float*)d_in[1];   // (361,361,361)
  float* out = (float*)d_out;                 // (1,128,361,720)

  float* Xp  = (float*)d_ws;                              // 361*364*256 f
  float* Y   = Xp + (size_t)MMAX * LPAD * NR;             // 46208*724   f
  float* W2p = Y + (size_t)ROWS2 * KPAD;                  // 362*736*2   f

  dim3 gT(LMAX, (MMAX + 31) / 32);            // (361, 12)
  k_pack_x<<<gT, 256, 0, stream>>>(x, Xp);

  int ptot = XPAD_N + YPAD_N;
  k_prep<<<(ptot + 255) / 256, 256, 0, stream>>>(Xp, Y);

  int wtot = (KPAD / 2) * JPAD;
  k_build_w<<<(wtot + 255) / 256, 256, 0, stream>>>(W2p);

  dim3 g1(MMAX, (NLAT + 31) / 32);            // (361, 12) x 8 waves (row tiles)
  k_legendre<<<g1, 256, 0, stream>>>(Xp, pct, Y);

  dim3 g2((NLON + 31) / 32, (ROWS2 / 32 + 7) / 8);  // (23, 181)
  k_ifft<<<g2, 256, 0, stream>>>(Y, W2p, out);
}